// LogLinearMamba2_62216896250126
// MI455X (gfx1250) — compile-verified
//
#include <hip/hip_runtime.h>
#include <hip/hip_bf16.h>
#include <math.h>

// Problem constants (match reference)
#define T_SEQ    1024
#define HEADS    32
#define HEADDIM  64
#define HIDDEN   1024
#define DSTATE   128
#define KCONV    4
#define NLVL     15
#define DIM      (HEADS * HEADDIM)                    // 2048
#define CONV_DIM (DIM + 2 * DSTATE)                   // 2304
#define PROJ     (DIM + CONV_DIM + HEADS * (NLVL+1))  // 4864
#define EPSV     1e-5f

typedef __attribute__((ext_vector_type(2))) float v2f;
typedef __attribute__((ext_vector_type(8))) float v8f;

__device__ __forceinline__ float softplusf(float x) {
  return (x > 0.f) ? (x + log1pf(__expf(-x))) : log1pf(__expf(x));
}
__device__ __forceinline__ float siluf(float x) {
  return x / (1.f + __expf(-x));
}

// ---------------------------------------------------------------------------
// f32 WMMA GEMM:  C[m,n] = sum_k A[m*LDA+k] * B[n*LDB+k] (+ bias[n])
// One wave per 16x64 output tile (4 accumulators share each A fragment).
// Strides are template constants so b-row offsets fold into the 24-bit
// instruction offset (no per-iteration 64-bit address adds), and the K-loop
// is software-pipelined (next iteration's 5 fragments are loaded before the
// current 4 WMMAs issue) so the scheduler can stagger s_wait_loadcnt.
// A-frag (16x4, ISA 7.12.2): lane L -> M = L%16, VGPR r -> K = 2*(L/16)+r.
// B-frag (4x16) symmetric:   lane L -> N = L%16, VGPR r -> K = 2*(L/16)+r.
// C/D frag: lane L, VGPR r -> M = r + 8*(L/16), N = L%16.
// ---------------------------------------------------------------------------
template <int LDA, int LDB, int LDC>
__global__ __launch_bounds__(256) void wmma_gemm_nt(
    const float* __restrict__ A, const float* __restrict__ B,
    const float* __restrict__ bias, float* __restrict__ C,
    int M, int N, int Kdim) {
  const int lane  = threadIdx.x & 31;
  const int wave  = (blockIdx.x * blockDim.x + threadIdx.x) >> 5;
  const int tiles_n = N >> 6;                    // 64-wide N tiles
  const int tm = wave / tiles_n;
  const int tn = wave - tm * tiles_n;
  if (tm >= (M >> 4)) return;                    // uniform per wave; EXEC stays full

  const int half = lane >> 4;
  const int idx  = lane & 15;
  const float* arow = A + (long)((tm << 4) + idx) * LDA + 2 * half;
  const float* brow = B + (long)((tn << 6) + idx) * LDB + 2 * half;

  v8f acc0 = {}, acc1 = {}, acc2 = {}, acc3 = {};

  // Prologue: load fragments for k = 0
  v2f a  = *(const v2f*)(arow);
  v2f b0 = *(const v2f*)(brow);
  v2f b1 = *(const v2f*)(brow + 16L * LDB);
  v2f b2 = *(const v2f*)(brow + 32L * LDB);
  v2f b3 = *(const v2f*)(brow + 48L * LDB);

  for (int k = 4; k < Kdim; k += 4) {
    // Prefetch next iteration's fragments (independent of current WMMAs)
    const v2f an  = *(const v2f*)(arow + k);
    const v2f b0n = *(const v2f*)(brow + k);
    const v2f b1n = *(const v2f*)(brow + k + 16L * LDB);
    const v2f b2n = *(const v2f*)(brow + k + 32L * LDB);
    const v2f b3n = *(const v2f*)(brow + k + 48L * LDB);
    acc0 = __builtin_amdgcn_wmma_f32_16x16x4_f32(false, a, false, b0, (short)0, acc0, false, false);
    acc1 = __builtin_amdgcn_wmma_f32_16x16x4_f32(false, a, false, b1, (short)0, acc1, false, false);
    acc2 = __builtin_amdgcn_wmma_f32_16x16x4_f32(false, a, false, b2, (short)0, acc2, false, false);
    acc3 = __builtin_amdgcn_wmma_f32_16x16x4_f32(false, a, false, b3, (short)0, acc3, false, false);
    a = an; b0 = b0n; b1 = b1n; b2 = b2n; b3 = b3n;
  }
  // Epilogue WMMAs for the last K-step
  acc0 = __builtin_amdgcn_wmma_f32_16x16x4_f32(false, a, false, b0, (short)0, acc0, false, false);
  acc1 = __builtin_amdgcn_wmma_f32_16x16x4_f32(false, a, false, b1, (short)0, acc1, false, false);
  acc2 = __builtin_amdgcn_wmma_f32_16x16x4_f32(false, a, false, b2, (short)0, acc2, false, false);
  acc3 = __builtin_amdgcn_wmma_f32_16x16x4_f32(false, a, false, b3, (short)0, acc3, false, false);

  const int col = (tn << 6) + idx;
  const float bc0 = bias ? bias[col]      : 0.f;
  const float bc1 = bias ? bias[col + 16] : 0.f;
  const float bc2 = bias ? bias[col + 32] : 0.f;
  const float bc3 = bias ? bias[col + 48] : 0.f;
#pragma unroll
  for (int r = 0; r < 8; ++r) {
    const long m = (tm << 4) + r + 8 * half;
    float* crow = C + m * LDC + col;
    crow[0]  = acc0[r] + bc0;
    crow[16] = acc1[r] + bc1;
    crow[32] = acc2[r] + bc2;
    crow[48] = acc3[r] + bc3;
  }
}

// ---------------------------------------------------------------------------
// Depthwise causal conv (K=4) + SiLU over the xBC slice of zxbcdtdl.
// ---------------------------------------------------------------------------
__global__ void conv_silu_kernel(const float* __restrict__ zx,
                                 const float* __restrict__ cw,
                                 float* __restrict__ xbc) {
  const int i = blockIdx.x * blockDim.x + threadIdx.x;
  if (i >= T_SEQ * CONV_DIM) return;
  const int t = i / CONV_DIM;
  const int c = i - t * CONV_DIM;
  float acc = 0.f;
#pragma unroll
  for (int j = 0; j < KCONV; ++j) {
    const int ts = t - (KCONV - 1) + j;
    if (ts >= 0) acc += zx[(long)ts * PROJ + DIM + c] * cw[c * KCONV + j];
  }
  xbc[i] = siluf(acc);
}

// ---------------------------------------------------------------------------
// dt = softplus(dt_raw + dt_bias), g = -exp(A_log)*dt, lam = softplus(L*dl)
// ---------------------------------------------------------------------------
__global__ void dt_lam_kernel(const float* __restrict__ zx,
                              const float* __restrict__ dtb,
                              const float* __restrict__ Alog,
                              const float* __restrict__ Lm,
                              float* __restrict__ dtsp,
                              float* __restrict__ g,
                              float* __restrict__ lam) {
  const int i = blockIdx.x * blockDim.x + threadIdx.x;
  if (i >= T_SEQ * HEADS) return;
  const int t = i / HEADS;
  const int h = i - t * HEADS;
  const float d = softplusf(zx[(long)t * PROJ + DIM + CONV_DIM + h] + dtb[h]);
  dtsp[i] = d;
  g[i]    = -__expf(Alog[h]) * d;
  const float* dl = zx + (long)t * PROJ + DIM + CONV_DIM + HEADS + h * NLVL;
#pragma unroll
  for (int l = 0; l < NLVL; ++l)
    lam[(long)i * NLVL + l] = softplusf(Lm[h * NLVL + l] * dl[l]);
}

// gc[h][t] = cumsum_t g[t][h]   (32 serial scans; trivial cost)
__global__ void cumsum_kernel(const float* __restrict__ g, float* __restrict__ gc) {
  const int h = threadIdx.x;
  if (h >= HEADS) return;
  float acc = 0.f;
  for (int t = 0; t < T_SEQ; ++t) {
    acc += g[t * HEADS + h];
    gc[h * T_SEQ + t] = acc;
  }
}

// ---------------------------------------------------------------------------
// Attention: y[t,h,p] = sum_{s<=t} qk[t,s]*exp(gc[h,t]-gc[h,s])*lam[t,h,lvl(t,s)]
//                       * v[s,h,p]   + x[t,h,p]*D[h]
// lvl(t,s) = 0 if s==t else 32-clz(t^s)  (Fenwick closed form).
// One wave per (head, 16-row t-tile); 4 WMMA accumulators cover p=0..63.
// ---------------------------------------------------------------------------
__global__ __launch_bounds__(256) void attn_kernel(
    const float* __restrict__ qk, const float* __restrict__ gc,
    const float* __restrict__ lam, const float* __restrict__ xbc,
    const float* __restrict__ dtsp, const float* __restrict__ Dv,
    float* __restrict__ y) {
  const int wave = (blockIdx.x * blockDim.x + threadIdx.x) >> 5;
  const int lane = threadIdx.x & 31;
  const int h  = wave >> 6;     // 64 t-tiles per head
  const int tt = wave & 63;
  const int t0 = tt << 4;
  const int half = lane >> 4;
  const int idx  = lane & 15;
  const int trow = t0 + idx;

  const float gct = gc[h * T_SEQ + trow];
  const float* lam_row = lam + ((long)trow * HEADS + h) * NLVL;

  v8f acc0 = {}, acc1 = {}, acc2 = {}, acc3 = {};
  const int smax = t0 + 16;                 // covers the diagonal tile
  for (int s = 0; s < smax; s += 4) {
    const int sr0 = s + 2 * half;
    // A fragment: att[trow, sr0+r]
    v2f a;
#pragma unroll
    for (int r = 0; r < 2; ++r) {
      const int sc = sr0 + r;
      float av = 0.f;
      if (sc <= trow) {
        const float dec = __expf(gct - gc[h * T_SEQ + sc]);
        const int lvl = (sc == trow) ? 0 : (32 - __clz(trow ^ sc));
        av = qk[(long)trow * T_SEQ + sc] * dec * lam_row[lvl];
      }
      a[r] = av;
    }
    // B fragments: v[sr0+r, h, p] = x * dt
    const float d0 = dtsp[sr0 * HEADS + h];
    const float d1 = dtsp[(sr0 + 1) * HEADS + h];
    const float* vr0 = xbc + (long)sr0 * CONV_DIM + h * HEADDIM;
    const float* vr1 = vr0 + CONV_DIM;
    v2f b;
    b[0] = vr0[idx]      * d0;  b[1] = vr1[idx]      * d1;
    acc0 = __builtin_amdgcn_wmma_f32_16x16x4_f32(false, a, false, b, (short)0, acc0, false, false);
    b[0] = vr0[16 + idx] * d0;  b[1] = vr1[16 + idx] * d1;
    acc1 = __builtin_amdgcn_wmma_f32_16x16x4_f32(false, a, false, b, (short)0, acc1, false, false);
    b[0] = vr0[32 + idx] * d0;  b[1] = vr1[32 + idx] * d1;
    acc2 = __builtin_amdgcn_wmma_f32_16x16x4_f32(false, a, false, b, (short)0, acc2, false, false);
    b[0] = vr0[48 + idx] * d0;  b[1] = vr1[48 + idx] * d1;
    acc3 = __builtin_amdgcn_wmma_f32_16x16x4_f32(false, a, false, b, (short)0, acc3, false, false);
  }

  const float Dh = Dv[h];
#pragma unroll
  for (int r = 0; r < 8; ++r) {
    const int m = t0 + r + 8 * half;
    const float* xr = xbc + (long)m * CONV_DIM + h * HEADDIM;
    float* yr = y + (long)m * DIM + h * HEADDIM;
    yr[idx]      = acc0[r] + xr[idx]      * Dh;
    yr[16 + idx] = acc1[r] + xr[16 + idx] * Dh;
    yr[32 + idx] = acc2[r] + xr[32 + idx] * Dh;
    yr[48 + idx] = acc3[r] + xr[48 + idx] * Dh;
  }
}

// ---------------------------------------------------------------------------
// yz = y * silu(z);  yn = yz * rsqrt(mean(yz^2)+eps) * norm_w
// ---------------------------------------------------------------------------
__global__ void gate_norm_kernel(const float* __restrict__ y,
                                 const float* __restrict__ zx,
                                 const float* __restrict__ nw,
                                 float* __restrict__ yn) {
  const int t = blockIdx.x;
  __shared__ float red[256];
  float p = 0.f;
  for (int d = threadIdx.x; d < DIM; d += blockDim.x) {
    const float z = zx[(long)t * PROJ + d];
    const float v = y[(long)t * DIM + d] * siluf(z);
    yn[(long)t * DIM + d] = v;
    p += v * v;
  }
  red[threadIdx.x] = p;
  __syncthreads();
  for (int off = 128; off > 0; off >>= 1) {
    if ((int)threadIdx.x < off) red[threadIdx.x] += red[threadIdx.x + off];
    __syncthreads();
  }
  const float scale = rsqrtf(red[0] / (float)DIM + EPSV);
  for (int d = threadIdx.x; d < DIM; d += blockDim.x)
    yn[(long)t * DIM + d] = yn[(long)t * DIM + d] * scale * nw[d];
}

// ---------------------------------------------------------------------------
extern "C" void kernel_launch(void* const* d_in, const int* in_sizes, int n_in,
                              void* d_out, int out_size, void* d_ws, size_t ws_size,
                              hipStream_t stream) {
  (void)in_sizes; (void)n_in; (void)out_size; (void)ws_size;
  const float* hs   = (const float*)d_in[0];   // (T, HIDDEN)
  const float* Wp   = (const float*)d_in[1];   // (PROJ, HIDDEN)
  const float* bp   = (const float*)d_in[2];   // (PROJ,)
  const float* cw   = (const float*)d_in[3];   // (CONV_DIM, K)
  const float* dtb  = (const float*)d_in[4];   // (H,)
  const float* Alog = (const float*)d_in[5];   // (H,)
  const float* Lm   = (const float*)d_in[6];   // (H, NL)
  const float* Dv   = (const float*)d_in[7];   // (H,)
  const float* nw   = (const float*)d_in[8];   // (DIM,)
  const float* Wo   = (const float*)d_in[9];   // (HIDDEN, DIM)
  const float* bo   = (const float*)d_in[10];  // (HIDDEN,)
  float* out = (float*)d_out;                  // (T, HIDDEN) f32

  // Workspace carve-up (all f32, ~50.3 MB total)
  float* ws   = (float*)d_ws;
  float* zx   = ws;                                   // T*PROJ
  float* xbc  = zx   + (size_t)T_SEQ * PROJ;          // T*CONV_DIM
  float* dtsp = xbc  + (size_t)T_SEQ * CONV_DIM;      // T*H
  float* gbuf = dtsp + (size_t)T_SEQ * HEADS;         // T*H
  float* gcb  = gbuf + (size_t)T_SEQ * HEADS;         // H*T
  float* lamb = gcb  + (size_t)HEADS * T_SEQ;         // T*H*NL
  float* qkb  = lamb + (size_t)T_SEQ * HEADS * NLVL;  // T*T
  float* yb   = qkb  + (size_t)T_SEQ * T_SEQ;         // T*DIM
  float* ynb  = yb   + (size_t)T_SEQ * DIM;           // T*DIM

  // 1. in_proj GEMM: zx = hs @ Wp^T + bp      (1024 x 4864, K=1024)
  {
    const int tiles = (T_SEQ / 16) * (PROJ / 64);     // 4864 waves, 8 per block
    wmma_gemm_nt<HIDDEN, HIDDEN, PROJ><<<tiles / 8, 256, 0, stream>>>(
        hs, Wp, bp, zx, T_SEQ, PROJ, HIDDEN);
  }
  // 2. conv + silu
  conv_silu_kernel<<<(T_SEQ * CONV_DIM + 255) / 256, 256, 0, stream>>>(zx, cw, xbc);
  // 3. dt / g / lam
  dt_lam_kernel<<<(T_SEQ * HEADS + 255) / 256, 256, 0, stream>>>(zx, dtb, Alog, Lm,
                                                                 dtsp, gbuf, lamb);
  // 4. gc = cumsum(g)
  cumsum_kernel<<<1, 32, 0, stream>>>(gbuf, gcb);
  // 5. qk = C @ B^T   (1024 x 1024, K=128); C at offset DIM+DSTATE, B at DIM
  {
    const int tiles = (T_SEQ / 16) * (T_SEQ / 64);    // 1024
    wmma_gemm_nt<CONV_DIM, CONV_DIM, T_SEQ><<<tiles / 8, 256, 0, stream>>>(
        xbc + DIM + DSTATE, xbc + DIM, nullptr, qkb, T_SEQ, T_SEQ, DSTATE);
  }
  // 6. attention + D residual
  attn_kernel<<<(HEADS * (T_SEQ / 16)) / 8, 256, 0, stream>>>(qkb, gcb, lamb, xbc,
                                                              dtsp, Dv, yb);
  // 7. gate + RMS norm
  gate_norm_kernel<<<T_SEQ, 256, 0, stream>>>(yb, zx, nw, ynb);
  // 8. out_proj GEMM: out = yn @ Wo^T + bo   (1024 x 1024, K=2048)
  {
    const int tiles = (T_SEQ / 16) * (HIDDEN / 64);   // 1024
    wmma_gemm_nt<DIM, DIM, HIDDEN><<<tiles / 8, 256, 0, stream>>>(
        ynb, Wo, bo, out, T_SEQ, HIDDEN, DIM);
  }
}